// ClasswiseECE_11424613007597
// MI455X (gfx1250) — compile-verified
//
#include <hip/hip_runtime.h>
#include <hip/hip_bf16.h>
#include <stdint.h>

#define C_CLASSES 100
#define NBINS 15
#define NSEG (C_CLASSES * NBINS)          // 1500 segments
#define TILE_ROWS 64
#define BLOCK_THREADS 256
#define WAVES_PER_BLOCK (BLOCK_THREADS / 32)
#define ROWS_PER_WAVE (TILE_ROWS / WAVES_PER_BLOCK)   // 8
#define MAX_HIST_BLOCKS 2048

typedef __attribute__((ext_vector_type(4))) unsigned int v4u;
typedef __attribute__((ext_vector_type(8))) int          v8i;
typedef __attribute__((ext_vector_type(4))) int          v4i;

__device__ __forceinline__ float wave_max(float v) {
#pragma unroll
  for (int m = 16; m; m >>= 1) v = fmaxf(v, __shfl_xor(v, m, 32));
  return v;
}
__device__ __forceinline__ float wave_sum(float v) {
#pragma unroll
  for (int m = 16; m; m >>= 1) v += __shfl_xor(v, m, 32);
  return v;
}

// Issue a TDM 2D tile load: TILE_ROWS x C_CLASSES f32, contiguous rows,
// global -> LDS.  tensor_dim1 = rows_remaining so trailing OOB rows read as 0.
__device__ __forceinline__ void tdm_load_tile(const float* gptr,
                                              unsigned lds_off,
                                              unsigned rows_remaining) {
  unsigned long long ga = (unsigned long long)(uintptr_t)gptr;
  v4u g0;
  g0[0] = 1u;                                       // count=1 (valid), user mode
  g0[1] = lds_off;                                  // lds_addr (bytes)
  g0[2] = (unsigned)(ga & 0xffffffffull);           // global_addr[31:0]
  g0[3] = (unsigned)((ga >> 32) & 0x01ffffffull)    // global_addr[56:32]
          | (2u << 30);                             // type = 2 ("image")
  const unsigned dim0    = C_CLASSES;               // tensor dim0 (elements)
  const unsigned dim1    = rows_remaining;          // tensor dim1 (rows, OOB clamp)
  const unsigned tile0   = C_CLASSES;               // tile dim0
  const unsigned tile1   = TILE_ROWS;               // tile dim1
  const unsigned stride0 = C_CLASSES;               // row stride (elements)
  v8i g1;
  g1[0] = (int)(2u << 16);                               // data_size=4B, no flags
  g1[1] = (int)((dim0 & 0xffffu) << 16);                 // tensor_dim0[15:0]
  g1[2] = (int)((dim0 >> 16) | ((dim1 & 0xffffu) << 16));// dim0 hi | dim1 lo
  g1[3] = (int)((dim1 >> 16) | (tile0 << 16));           // dim1 hi | tile_dim0
  g1[4] = (int)(tile1);                                  // tile_dim1 (tile_dim2=0)
  g1[5] = (int)(stride0);                                // tensor_dim0_stride lo
  g1[6] = 0;                                             // stride0 hi | dim1_stride
  g1[7] = 0;
  v4i z4; z4[0] = 0; z4[1] = 0; z4[2] = 0; z4[3] = 0;
  v8i z8; z8[0] = 0; z8[1] = 0; z8[2] = 0; z8[3] = 0;
  z8[4] = 0; z8[5] = 0; z8[6] = 0; z8[7] = 0;
  // 6-arg form (clang-23 / therock-10.0 headers): groups 0..3 + extra group + cpol
  __builtin_amdgcn_tensor_load_to_lds(g0, g1, z4, z4, z8, 0);
}

__global__ void sce_zero_ws(float* ws) {
  int i = blockIdx.x * blockDim.x + threadIdx.x;
  if (i < 3 * NSEG) ws[i] = 0.0f;
}

__global__ __launch_bounds__(BLOCK_THREADS) void sce_hist(
    const float* __restrict__ logits, const int* __restrict__ labels,
    float* __restrict__ gcnt, float* __restrict__ gconf,
    float* __restrict__ gacc, int N) {
  __shared__ float    tile[2][TILE_ROWS * C_CLASSES];  // 51.2 KB double buffer
  __shared__ unsigned cntL[NSEG];
  __shared__ float    confL[NSEG];
  __shared__ float    accL[NSEG];

  const int tid  = threadIdx.x;
  const int lane = tid & 31;
  const int wid  = tid >> 5;

  for (int i = tid; i < NSEG; i += BLOCK_THREADS) {
    cntL[i] = 0u; confL[i] = 0.0f; accL[i] = 0.0f;
  }

  const int  total_tiles = (N + TILE_ROWS - 1) / TILE_ROWS;
  const bool leader      = (__builtin_amdgcn_readfirstlane(wid) == 0);

  int cur = 0;
  int t0  = blockIdx.x;
  if (t0 < total_tiles && leader) {
    tdm_load_tile(logits + (size_t)t0 * TILE_ROWS * C_CLASSES,
                  (unsigned)(uintptr_t)&tile[0][0],
                  (unsigned)(N - t0 * TILE_ROWS));
  }

  for (int t = t0; t < total_tiles; t += gridDim.x) {
    __builtin_amdgcn_s_wait_tensorcnt(0);   // SOPP: no-op for non-issuing waves
    __syncthreads();                         // tile[cur] visible to all waves

    int tn = t + gridDim.x;
    if (tn < total_tiles && leader) {
      tdm_load_tile(logits + (size_t)tn * TILE_ROWS * C_CLASSES,
                    (unsigned)(uintptr_t)&tile[cur ^ 1][0],
                    (unsigned)(N - tn * TILE_ROWS));
    }

    // One row per wave; lane covers classes lane, lane+32, lane+64, lane+96.
    for (int r = 0; r < ROWS_PER_WAVE; ++r) {
      const int rit  = wid * ROWS_PER_WAVE + r;
      const int grow = t * TILE_ROWS + rit;
      if (grow >= N) break;                  // wave-uniform, no barriers inside
      const float* rowp = &tile[cur][rit * C_CLASSES];

      float x[4];
      float m = -3.0e38f;
#pragma unroll
      for (int k = 0; k < 4; ++k) {
        const int c = lane + 32 * k;
        x[k] = (c < C_CLASSES) ? rowp[c] : -3.0e38f;
        m = fmaxf(m, x[k]);
      }
      m = wave_max(m);

      float e[4];
      float s = 0.0f;
#pragma unroll
      for (int k = 0; k < 4; ++k) {
        const int c = lane + 32 * k;
        e[k] = (c < C_CLASSES) ? __expf(x[k] - m) : 0.0f;
        s += e[k];
      }
      s = wave_sum(s);
      const float inv = 1.0f / s;
      const int   lab = labels[grow];

#pragma unroll
      for (int k = 0; k < 4; ++k) {
        const int c = lane + 32 * k;
        if (c < C_CLASSES) {
          const float p = e[k] * inv;
          int b = (int)__builtin_ceilf(p * (float)NBINS) - 1;
          b = b < 0 ? 0 : (b > NBINS - 1 ? NBINS - 1 : b);
          const int seg = c * NBINS + b;     // 32 lanes -> 32 distinct banks
          atomicAdd(&cntL[seg], 1u);
          atomicAdd(&confL[seg], p);
          if (c == lab) atomicAdd(&accL[seg], 1.0f);
        }
      }
    }
    __syncthreads();                         // done reading tile[cur]
    cur ^= 1;
  }

  __syncthreads();
  for (int i = tid; i < NSEG; i += BLOCK_THREADS) {
    const unsigned c  = cntL[i];
    const float    cf = confL[i];
    const float    af = accL[i];
    if (c)          unsafeAtomicAdd(&gcnt[i], (float)c);
    if (cf != 0.0f) unsafeAtomicAdd(&gconf[i], cf);
    if (af != 0.0f) unsafeAtomicAdd(&gacc[i], af);
  }
}

__global__ __launch_bounds__(256) void sce_final(
    const float* __restrict__ gcnt, const float* __restrict__ gconf,
    const float* __restrict__ gacc, float* __restrict__ out, int N) {
  __shared__ float red[8];
  float local = 0.0f;
  for (int i = threadIdx.x; i < NSEG; i += 256) {
    const float cnt = gcnt[i];
    if (cnt > 0.0f) {
      const float gap = fabsf((gconf[i] - gacc[i]) / cnt);
      local += gap * (cnt / (float)N);
    }
  }
  local = wave_sum(local);
  const int lane = threadIdx.x & 31;
  const int wid  = threadIdx.x >> 5;
  if (lane == 0) red[wid] = local;
  __syncthreads();
  if (wid == 0) {
    float v = (lane < 8) ? red[lane] : 0.0f;
    v = wave_sum(v);
    if (lane == 0) out[0] = v / (float)C_CLASSES;
  }
}

extern "C" void kernel_launch(void* const* d_in, const int* in_sizes, int n_in,
                              void* d_out, int out_size, void* d_ws,
                              size_t ws_size, hipStream_t stream) {
  const float* logits = (const float*)d_in[0];
  const int*   labels = (const int*)d_in[1];
  const int    N      = in_sizes[1];          // 500000 rows; C fixed at 100

  float* ws    = (float*)d_ws;                // 3*1500 floats = 18 KB scratch
  float* gcnt  = ws;
  float* gconf = ws + NSEG;
  float* gacc  = ws + 2 * NSEG;

  sce_zero_ws<<<(3 * NSEG + 255) / 256, 256, 0, stream>>>(ws);

  const int total_tiles = (N + TILE_ROWS - 1) / TILE_ROWS;
  const int grid = total_tiles < MAX_HIST_BLOCKS ? total_tiles : MAX_HIST_BLOCKS;
  sce_hist<<<grid, BLOCK_THREADS, 0, stream>>>(logits, labels, gcnt, gconf,
                                               gacc, N);

  sce_final<<<1, 256, 0, stream>>>(gcnt, gconf, gacc, (float*)d_out, N);
}